// Tpe_28527172780147
// MI455X (gfx1250) — compile-verified
//
#include <hip/hip_runtime.h>
#include <hip/hip_bf16.h>
#include <math.h>

typedef float v2f __attribute__((ext_vector_type(2)));
typedef float v8f __attribute__((ext_vector_type(8)));

#define HEADS   16
#define DHEAD   64
#define LSEQ    128     // chunk length for the time scan
#define NCHUNK  32      // 4096 / 128
#define EPSRMS  1e-6f

__device__ __forceinline__ float sigmoidf_dev(float a) {
    return 1.0f / (1.0f + expf(-a));
}

// -----------------------------------------------------------------------------
// Kernel 1: chunk-local decay scan.  y_t = lam_h * y_{t-1} + x_t  within chunk.
// -----------------------------------------------------------------------------
__global__ __launch_bounds__(256) void scan_local(
    const float* __restrict__ x, const float* __restrict__ log_decay,
    float* __restrict__ ylocal, float* __restrict__ echunk, int N, int D)
{
    const int b = blockIdx.x / NCHUNK;
    const int c = blockIdx.x % NCHUNK;
    const int tid = threadIdx.x;

    float y[4], lam[4];
#pragma unroll
    for (int j = 0; j < 4; ++j) {
        int d = tid + 256 * j;
        lam[j] = sigmoidf_dev(log_decay[d >> 6]);
        y[j] = 0.0f;
    }
    const float* xp = x + ((size_t)b * N + (size_t)c * LSEQ) * D;
    float* yp = ylocal + ((size_t)b * N + (size_t)c * LSEQ) * D;

    for (int i = 0; i < LSEQ; ++i) {
#pragma unroll
        for (int j = 0; j < 4; ++j) {
            int d = tid + 256 * j;
            float xv = xp[(size_t)i * D + d];
            y[j] = fmaf(lam[j], y[j], xv);
            yp[(size_t)i * D + d] = y[j];
        }
    }
#pragma unroll
    for (int j = 0; j < 4; ++j)
        echunk[((size_t)(b * NCHUNK + c)) * D + tid + 256 * j] = y[j];
}

// -----------------------------------------------------------------------------
// Kernel 2: carry scan across chunks: carry_c = lam^L * carry_{c-1} + e_{c-1}
// -----------------------------------------------------------------------------
__global__ __launch_bounds__(256) void scan_carry(
    const float* __restrict__ log_decay, const float* __restrict__ echunk,
    float* __restrict__ carry, int D)
{
    const int idx = blockIdx.x * blockDim.x + threadIdx.x; // 0 .. B*D-1
    const int b = idx / D;
    const int d = idx % D;
    float lam = sigmoidf_dev(log_decay[d >> 6]);
    float lamL = lam;
#pragma unroll
    for (int i = 0; i < 7; ++i) lamL *= lamL;   // lam^128
    float cy = 0.0f;
    for (int c = 0; c < NCHUNK; ++c) {
        size_t off = ((size_t)(b * NCHUNK + c)) * D + d;
        carry[off] = cy;
        cy = fmaf(lamL, cy, echunk[off]);
    }
}

// -----------------------------------------------------------------------------
// Kernel 3: per-head constant c_h = dot(qh, kh). One wave32 per head.
// -----------------------------------------------------------------------------
__global__ void head_dots(const float* __restrict__ q, const float* __restrict__ k,
                          float* __restrict__ ch)
{
    const int wave = threadIdx.x >> 5;
    const int lane = threadIdx.x & 31;
    const int base = wave * DHEAD;
    float s = q[base + lane] * k[base + lane] +
              q[base + 32 + lane] * k[base + 32 + lane];
#pragma unroll
    for (int off = 16; off > 0; off >>= 1)
        s += __shfl_down(s, off, 32);
    if (lane == 0) ch[wave] = s;
}

// -----------------------------------------------------------------------------
// Kernel 4: apply carry, scale by c_h, fused RMSNorm. One block per (b, t) row.
// -----------------------------------------------------------------------------
__global__ __launch_bounds__(256) void finalize_norm(
    const float* __restrict__ carry, const float* __restrict__ log_decay,
    const float* __restrict__ ch, const float* __restrict__ nw,
    float* __restrict__ Ybuf, int N, int D)
{
    const int row = blockIdx.x;           // b*N + t
    const int b = row / N;
    const int t = row % N;
    const int c = t / LSEQ;
    const int i = t % LSEQ;
    const int tid = threadIdx.x;

    float* yr = Ybuf + (size_t)row * D;
    const float* cr = carry + ((size_t)(b * NCHUNK + c)) * D;

    float vals[4];
    float ss = 0.0f;
#pragma unroll
    for (int j = 0; j < 4; ++j) {
        int d = tid + 256 * j;
        int h = d >> 6;
        float lam = sigmoidf_dev(log_decay[h]);
        int e = i + 1;
        float p = 1.0f, base = lam;
        while (e) { if (e & 1) p *= base; base *= base; e >>= 1; }
        float v = ch[h] * fmaf(p, cr[d], yr[d]);
        vals[j] = v;
        ss += v * v;
    }
    __shared__ float red[256];
    red[tid] = ss;
    __syncthreads();
    for (int s2 = 128; s2 > 0; s2 >>= 1) {
        if (tid < s2) red[tid] += red[tid + s2];
        __syncthreads();
    }
    const float scale = rsqrtf(red[0] / (float)D + EPSRMS);
#pragma unroll
    for (int j = 0; j < 4; ++j) {
        int d = tid + 256 * j;
        yr[d] = vals[j] * scale * nw[d];
    }
}

// -----------------------------------------------------------------------------
// Kernel 5: fp32 WMMA GEMM with CDNA5 async global->LDS double buffering.
// C[m,n] = sum_k A[m,k] * Bw[n,k]   (NT gemm)
// -----------------------------------------------------------------------------
#define BM 128
#define BN 128
#define BK 32
#define LP 36   // padded LDS row stride in floats (bank-conflict-free frags)

__device__ __forceinline__ void async_load_b128(uint32_t lds_off, const float* gaddr)
{
    // GLOBAL_LOAD_ASYNC_TO_LDS_B128: memory -> LDS, tracked by ASYNCcnt.
    asm volatile("global_load_async_to_lds_b128 %0, %1, off"
                 :: "v"(lds_off), "v"(gaddr)
                 : "memory");
}
__device__ __forceinline__ void wait_async_le8()  { asm volatile("s_wait_asynccnt 0x8" ::: "memory"); }
__device__ __forceinline__ void wait_async_le0()  { asm volatile("s_wait_asynccnt 0x0" ::: "memory"); }

__global__ __launch_bounds__(256) void gemm_f32_wmma(
    const float* __restrict__ A, const float* __restrict__ Bw,
    float* __restrict__ C, int M, int Nn, int K)
{
    __shared__ float As[2][BM][LP];
    __shared__ float Bs[2][BM][LP];

    const int tid  = threadIdx.x;
    const int wave = tid >> 5;     // 0..7
    const int lane = tid & 31;
    const int lo   = lane & 15;
    const int hi   = lane >> 4;
    const int wm   = wave >> 1;    // 0..3 : 32-row M subtile
    const int wn   = wave & 1;     // 0..1 : 64-col N subtile
    const int mBlock = blockIdx.x * BM;
    const int nBlock = blockIdx.y * BN;

    v8f acc[2][4];
#pragma unroll
    for (int mi = 0; mi < 2; ++mi)
#pragma unroll
        for (int ni = 0; ni < 4; ++ni)
            acc[mi][ni] = (v8f)(0.0f);

    const int sRow = tid >> 3;        // 0..31
    const int sCol = (tid & 7) * 4;   // 0..28

    // LDS byte offsets of the staging buffers (generic-pointer low 32 bits ==
    // LDS offset per the CDNA5 shared-aperture rule).
    const uint32_t ldsA[2] = { (uint32_t)(uintptr_t)&As[0][0][0],
                               (uint32_t)(uintptr_t)&As[1][0][0] };
    const uint32_t ldsB[2] = { (uint32_t)(uintptr_t)&Bs[0][0][0],
                               (uint32_t)(uintptr_t)&Bs[1][0][0] };

    const float* Abase = A  + (size_t)mBlock * K;
    const float* Bbase = Bw + (size_t)nBlock * K;

    // Issue one stage: 4 A-rows + 4 B-rows of b128 per thread (8 async ops).
    auto issue_stage = [&](int kb, int buf) {
#pragma unroll
        for (int p = 0; p < 4; ++p) {
            int r = sRow + 32 * p;
            uint32_t loff = (uint32_t)((r * LP + sCol) * 4);
            async_load_b128(ldsA[buf] + loff, Abase + (size_t)r * K + kb + sCol);
            async_load_b128(ldsB[buf] + loff, Bbase + (size_t)r * K + kb + sCol);
        }
    };

    issue_stage(0, 0);   // prologue: stage 0 -> buffer 0

    const int nSteps = K / BK;
    for (int s = 0; s < nSteps; ++s) {
        const int cur = s & 1;
        // Prefetch next K-block into the other buffer (fully consumed last
        // iteration -- trailing barrier below guarantees it).
        if (s + 1 < nSteps) {
            issue_stage((s + 1) * BK, cur ^ 1);
            wait_async_le8();    // own loads for 'cur' complete, prefetch in flight
        } else {
            wait_async_le0();
        }
        __syncthreads();         // all waves' 'cur' tiles resident in LDS

#pragma unroll
        for (int kk = 0; kk < BK; kk += 4) {
            v2f aF[2], bF[4];
#pragma unroll
            for (int mi = 0; mi < 2; ++mi)
                aF[mi] = *(const v2f*)&As[cur][wm * 32 + mi * 16 + lo][kk + 2 * hi];
#pragma unroll
            for (int ni = 0; ni < 4; ++ni)
                bF[ni] = *(const v2f*)&Bs[cur][wn * 64 + ni * 16 + lo][kk + 2 * hi];
#pragma unroll
            for (int mi = 0; mi < 2; ++mi)
#pragma unroll
                for (int ni = 0; ni < 4; ++ni)
                    acc[mi][ni] = __builtin_amdgcn_wmma_f32_16x16x4_f32(
                        false, aF[mi], false, bF[ni], (short)0, acc[mi][ni],
                        false, false);
        }
        __syncthreads();         // all waves done reading 'cur' -> reusable
    }

    // D-matrix layout: VGPR e -> row (e + 8*hi), col lo within 16x16 tile.
#pragma unroll
    for (int mi = 0; mi < 2; ++mi) {
#pragma unroll
        for (int ni = 0; ni < 4; ++ni) {
            int rbase = mBlock + wm * 32 + mi * 16 + 8 * hi;
            int cidx  = nBlock + wn * 64 + ni * 16 + lo;
#pragma unroll
            for (int e = 0; e < 8; ++e)
                C[(size_t)(rbase + e) * Nn + cidx] = acc[mi][ni][e];
        }
    }
}

// -----------------------------------------------------------------------------
// Launch
// -----------------------------------------------------------------------------
extern "C" void kernel_launch(void* const* d_in, const int* in_sizes, int n_in,
                              void* d_out, int out_size, void* d_ws, size_t ws_size,
                              hipStream_t stream)
{
    const float* x         = (const float*)d_in[0];  // (B, N, D)
    const float* q         = (const float*)d_in[1];  // (D,)
    const float* k         = (const float*)d_in[2];  // (D,)
    const float* log_decay = (const float*)d_in[3];  // (H,)
    const float* nw        = (const float*)d_in[4];  // (D,)
    const float* o_w       = (const float*)d_in[5];  // (D, D) row-major
    float* out = (float*)d_out;

    const int B = 4, N = 4096, D = 1024;

    float* ws    = (float*)d_ws;
    float* Ybuf  = ws;                                   // B*N*D  (scan -> normalized o)
    float* ech   = Ybuf + (size_t)B * N * D;             // B*NCHUNK*D chunk-end values
    float* carry = ech  + (size_t)B * NCHUNK * D;        // B*NCHUNK*D carries
    float* ch    = carry + (size_t)B * NCHUNK * D;       // H head constants

    scan_local<<<B * NCHUNK, 256, 0, stream>>>(x, log_decay, Ybuf, ech, N, D);
    scan_carry<<<(B * D) / 256, 256, 0, stream>>>(log_decay, ech, carry, D);
    head_dots<<<1, 32 * HEADS, 0, stream>>>(q, k, ch);
    finalize_norm<<<B * N, 256, 0, stream>>>(carry, log_decay, ch, nw, Ybuf, N, D);

    dim3 grid(B * N / BM, D / BN);
    gemm_f32_wmma<<<grid, 256, 0, stream>>>(Ybuf, o_w, out, B * N, D, D);
}